// ARC_25005299597496
// MI455X (gfx1250) — compile-verified
//
#include <hip/hip_runtime.h>
#include <hip/hip_bf16.h>
#include <math.h>

typedef __attribute__((ext_vector_type(2))) float v2f;
typedef __attribute__((ext_vector_type(8))) float v8f;
typedef __attribute__((ext_vector_type(4))) unsigned int u32x4;
typedef __attribute__((ext_vector_type(8))) unsigned int u32x8;

#define WMMA_F32X4(a, b, c) \
    __builtin_amdgcn_wmma_f32_16x16x4_f32(false, (a), false, (b), (short)0, (c), false, false)

namespace cfg {
constexpr int B   = 4;
constexpr int SEQ = 2048;
constexpr int S   = 128;
constexpr int L   = SEQ + 2 * S;   // 2304
constexpr int DI  = 512;
constexpr int DK  = 64;
constexpr int DV  = 64;
constexpr int NT  = L / 16;        // 144 tiles of 16 rows
constexpr float NEGV  = -1.0e30f;
constexpr float SCALE = 0.125f;    // 1/sqrt(64)
}

// ---------------------------------------------------------------------------
// TDM: DMA one 16x64 fp32 tile (4 KB, row stride 64 floats) into LDS.
// D# built per CDNA5 ISA §8.3/8.4; group0/group1 in SGPR tuples (all uniform).
// Tracked by TENSORcnt; waited with s_wait_tensorcnt.
// ---------------------------------------------------------------------------
__device__ __forceinline__ void tdm_load_tile_16x64(const float* gsrc,
                                                    unsigned lds_off) {
    unsigned long long ga = (unsigned long long)(uintptr_t)gsrc;
    u32x4 g0;
    g0[0] = 1u;                                          // count=1, no gather
    g0[1] = lds_off;                                     // lds_addr (bytes)
    g0[2] = (unsigned)(ga & 0xFFFFFFFFu);                // global_addr[31:0]
    g0[3] = (unsigned)((ga >> 32) & 0x01FFFFFFu)         // global_addr[56:32]
          | (2u << 30);                                  // type = 2 ("image")
    u32x8 g1;
    g1[0] = (2u << 16);          // workgroup_mask=0, data_size=2 (4 bytes)
    g1[1] = (64u << 16);         // tensor_dim0 = 64  (bits 79:48, low half)
    g1[2] = (16u << 16);         // dim0 hi=0 | tensor_dim1 = 16 (low half)
    g1[3] = (64u << 16);         // dim1 hi=0 | tile_dim0 = 64
    g1[4] = 16u;                 // tile_dim1 = 16, tile_dim2 = 0
    g1[5] = 64u;                 // tensor_dim0_stride = 64 (low 32)
    g1[6] = 0u;                  // stride0 hi | stride1 lo
    g1[7] = 0u;                  // stride1 hi
    asm volatile("tensor_load_to_lds %0, %1" :: "s"(g0), "s"(g1) : "memory");
}

// ---------------------------------------------------------------------------
// Kernel 1: per-segment LayerNorm fused with Q/K/V projections (512 -> 64)
// 1 block = 16 rows, 4 waves; each wave owns one 16-col strip of Q, K, V.
// ---------------------------------------------------------------------------
__global__ __launch_bounds__(128) void ARC_ln_proj(
    const float* __restrict__ x,
    const float* __restrict__ Wq,   const float* __restrict__ Wk,   const float* __restrict__ Wv,
    const float* __restrict__ Wq_s, const float* __restrict__ Wk_s, const float* __restrict__ Wv_s,
    const float* __restrict__ Wq_e, const float* __restrict__ Wk_e, const float* __restrict__ Wv_e,
    const float* __restrict__ g_m,  const float* __restrict__ b_m,
    const float* __restrict__ g_s,  const float* __restrict__ b_s,
    const float* __restrict__ g_e,  const float* __restrict__ b_e,
    float* __restrict__ qf, float* __restrict__ kf, float* __restrict__ vf)
{
    using namespace cfg;
    __shared__ float xs[16 * DI];     // 32 KB
    __shared__ float mv[16][2];

    const int tid = threadIdx.x;
    const int r0  = blockIdx.x * 16;  // global row over B*L (tiles never cross batch)
    const int lr  = r0 % L;           // row within batch

    // segment-uniform weight selection
    const float *pWq, *pWk, *pWv, *pg, *pb;
    if (lr < S)            { pWq = Wq_s; pWk = Wk_s; pWv = Wv_s; pg = g_s; pb = b_s; }
    else if (lr >= L - S)  { pWq = Wq_e; pWk = Wk_e; pWv = Wv_e; pg = g_e; pb = b_e; }
    else                   { pWq = Wq;   pWk = Wk;   pWv = Wv;   pg = g_m; pb = b_m; }

    // stage 16x512 rows into LDS (float4)
    const float4* xsrc = (const float4*)(x + (size_t)r0 * DI);
    float4* xdst = (float4*)xs;
    for (int i = tid; i < 16 * DI / 4; i += 128) xdst[i] = xsrc[i];
    __syncthreads();

    // mean / var: 8 lanes per row, shuffle-reduce within 8-lane segments
    {
        const int row = tid >> 3, sub = tid & 7;
        float s0 = 0.f, s1 = 0.f;
        for (int j = 0; j < DI / 8; ++j) {
            float v = xs[row * DI + sub + 8 * j];
            s0 += v; s1 += v * v;
        }
        for (int off = 1; off < 8; off <<= 1) {
            s0 += __shfl_xor(s0, off, 8);
            s1 += __shfl_xor(s1, off, 8);
        }
        if (sub == 0) {
            float m  = s0 * (1.f / DI);
            float va = s1 * (1.f / DI) - m * m;
            mv[row][0] = m;
            mv[row][1] = rsqrtf(va + 1e-5f);
        }
    }
    __syncthreads();

    // normalize in place
    for (int i = tid; i < 16 * DI; i += 128) {
        const int row = i >> 9, d = i & (DI - 1);
        xs[i] = (xs[i] - mv[row][0]) * mv[row][1] * pg[d] + pb[d];
    }
    __syncthreads();

    // WMMA f32 16x16x4: 128 K-steps, 3 output tiles per wave
    const int wv = tid >> 5, lane = tid & 31;
    const int rowA = lane & 15, hi = lane >> 4;
    const int col  = wv * 16 + rowA;

    v8f cq = {}, ck = {}, cv = {};
    for (int kk = 0; kk < DI / 4; ++kk) {
        const int d = kk * 4 + 2 * hi;
        v2f a;  a.x  = xs[rowA * DI + d];       a.y  = xs[rowA * DI + d + 1];
        v2f bq; bq.x = pWq[d * DK + col];       bq.y = pWq[(d + 1) * DK + col];
        v2f bk; bk.x = pWk[d * DK + col];       bk.y = pWk[(d + 1) * DK + col];
        v2f bv; bv.x = pWv[d * DV + col];       bv.y = pWv[(d + 1) * DV + col];
        cq = WMMA_F32X4(a, bq, cq);
        ck = WMMA_F32X4(a, bk, ck);
        cv = WMMA_F32X4(a, bv, cv);
    }
    for (int vg = 0; vg < 8; ++vg) {
        const int row = vg + 8 * hi;
        const size_t o = (size_t)(r0 + row) * DK + col;
        qf[o] = cq[vg]; kf[o] = ck[vg]; vf[o] = cv[vg];
    }
}

// ---------------------------------------------------------------------------
// Kernel 2: fused CoPE-bias flash attention, descending-k scan.
// 1 block = 1 wave = one 16-row Q tile. K/V tiles double-buffered in LDS via
// TDM tensor_load_to_lds. Raw QK^T tile (WMMA) feeds both the sigmoid-gate
// suffix scan and the masked/biased online softmax; P@V via WMMA.
// ---------------------------------------------------------------------------
__global__ __launch_bounds__(32) void ARC_cope_attn(
    const float* __restrict__ qf, const float* __restrict__ kf,
    const float* __restrict__ vf, const float* __restrict__ cope,
    float* __restrict__ out)
{
    using namespace cfg;
    const int qt = blockIdx.x % NT;
    const int b  = blockIdx.x / NT;
    const int q0 = qt * 16;
    const bool qmid = (q0 >= S) && (q0 < L - S);

    const int lane = threadIdx.x;
    const int lg = lane & 15, hi = lane >> 4;

    __shared__ float lint[16 * S];       // q @ cope_emb, 8 KB
    __shared__ float pbuf[16 * 18];      // P tile C-layout -> A-layout staging
    __shared__ float kbuf[2][16 * DK];   // TDM double buffer: K tiles (2x4 KB)
    __shared__ float vbuf[2][16 * DV];   // TDM double buffer: V tiles (2x4 KB)

    const unsigned koff0 = (unsigned)(uintptr_t)&kbuf[0][0];
    const unsigned koff1 = (unsigned)(uintptr_t)&kbuf[1][0];
    const unsigned voff0 = (unsigned)(uintptr_t)&vbuf[0][0];
    const unsigned voff1 = (unsigned)(uintptr_t)&vbuf[1][0];

    // Q tile A-fragments for all 16 K-steps (K=64)
    const float* qrow = qf + (size_t)(b * L + q0 + lg) * DK;
    v2f aq[16];
    for (int kk = 0; kk < 16; ++kk) {
        aq[kk].x = qrow[kk * 4 + 2 * hi];
        aq[kk].y = qrow[kk * 4 + 2 * hi + 1];
    }

    const int kt_caus  = qt;                               // causal top tile
    const int kt_start = qmid ? (L - S) / 16 - 1 : qt;     // gate suffix from middle top
    const float* kfb = kf + (size_t)b * L * DK;
    const float* vfb = vf + (size_t)b * L * DV;

    // Prime the TDM pipeline with tile kt_start into buffer 0.
    tdm_load_tile_16x64(kfb + (size_t)kt_start * 16 * DK, koff0);
    tdm_load_tile_16x64(vfb + (size_t)kt_start * 16 * DV, voff0);

    // CoPE interpolation table: lint[16][128] = Qtile @ cope_emb (WMMA),
    // overlapped with the first TDM transfer.
    if (qmid) {
        for (int pt = 0; pt < S / 16; ++pt) {
            v8f c = {};
            for (int kk = 0; kk < 16; ++kk) {
                const int d = kk * 4 + 2 * hi;
                v2f bc;
                bc.x = cope[d * S + pt * 16 + lg];
                bc.y = cope[(d + 1) * S + pt * 16 + lg];
                c = WMMA_F32X4(aq[kk], bc, c);
            }
            for (int vg = 0; vg < 8; ++vg)
                lint[(vg + 8 * hi) * S + pt * 16 + lg] = c[vg];
        }
    }
    __syncthreads();

    float m[8], l[8], suf[8];
    v8f acc[4];
    const v8f vzero = {};
    for (int i = 0; i < 8; ++i) { m[i] = -3.0e38f; l[i] = 0.f; suf[i] = 0.f; }
    for (int t = 0; t < 4; ++t) acc[t] = vzero;

    // All tiles kt_start..0 are needed (contiguous): middle-q tiles need gates
    // for every middle kt and scores for kt<=qt; edge-q tiles need kt<=qt.
    int p = 0;
    for (int kt = kt_start; kt >= 0; --kt, p ^= 1) {
        const int  k0      = kt * 16;
        const bool kmid    = (k0 >= S) && (k0 < L - S);
        const bool doScore = (kt <= kt_caus);
        const bool doGate  = qmid && kmid;

        // Issue next tile into the other buffer, then wait for current one.
        if (kt > 0) {
            const unsigned kn = p ? koff0 : koff1;
            const unsigned vn = p ? voff0 : voff1;
            tdm_load_tile_16x64(kfb + (size_t)(kt - 1) * 16 * DK, kn);
            tdm_load_tile_16x64(vfb + (size_t)(kt - 1) * 16 * DV, vn);
            __builtin_amdgcn_s_wait_tensorcnt(2);   // current tile done (in-order)
        } else {
            __builtin_amdgcn_s_wait_tensorcnt(0);
        }

        const float* kl = &kbuf[p][0];
        const float* vl = &vbuf[p][0];

        // raw scores: Qtile @ Ktile^T  (16x16, K=64), K from LDS
        v8f c = {};
        for (int kk = 0; kk < 16; ++kk) {
            v2f bb;
            bb.x = kl[lg * DK + kk * 4 + 2 * hi];
            bb.y = kl[lg * DK + kk * 4 + 2 * hi + 1];
            c = WMMA_F32X4(aq[kk], bb, c);
        }

        float bias[8];
        for (int i = 0; i < 8; ++i) bias[i] = 0.f;

        if (doGate) {
            // sigmoid gates, in-tile inclusive suffix scan over descending k,
            // running suffix across tiles, clamp, interpolate CoPE bias.
            for (int vg = 0; vg < 8; ++vg) {
                const float g = 1.f / (1.f + __expf(-c[vg]));
                float v = g;
                for (int off = 1; off < 16; off <<= 1) {
                    float t = __shfl_down(v, off, 16);
                    if (lg + off < 16) v += t;
                }
                const float tot = __shfl(v, 0, 16);       // whole-tile row sum
                float pos = suf[vg] + v;                  // inclusive suffix sum
                suf[vg] += tot;
                pos = fminf(pos, (float)(S - 1));
                const float pf = floorf(pos);
                const float w  = pos - pf;
                const int ip = (int)pf;
                const int ic = (int)ceilf(pos);
                const int row = vg + 8 * hi;
                bias[vg] = lint[row * S + ic] * w + lint[row * S + ip] * (1.f - w);
            }
        }
        if (!doScore) continue;

        // masked/scaled/biased scores -> online softmax update
        for (int vg = 0; vg < 8; ++vg) {
            const int gr = q0 + vg + 8 * hi;
            const int gc = k0 + lg;
            float s = c[vg] * SCALE + bias[vg];
            if (gc > gr) s += NEGV;                        // causal mask (ref: NEG+bias)
            float tm = s;
            for (int off = 1; off < 16; off <<= 1) tm = fmaxf(tm, __shfl_xor(tm, off, 16));
            const float mn = fmaxf(m[vg], tm);
            const float pe = __expf(s - mn);
            float rs = pe;
            for (int off = 1; off < 16; off <<= 1) rs += __shfl_xor(rs, off, 16);
            const float rescale = __expf(m[vg] - mn);
            l[vg] = l[vg] * rescale + rs;
            m[vg] = mn;
            for (int t = 0; t < 4; ++t) acc[t][vg] *= rescale;
            pbuf[(vg + 8 * hi) * 18 + lg] = pe;            // C-layout -> LDS
        }
        __syncthreads();

        // acc += P(16x16) @ Vtile(16x64), V from LDS, 4 col tiles x 4 K-steps
        for (int t = 0; t < 4; ++t) {
            v8f d = acc[t];
            for (int kk = 0; kk < 4; ++kk) {
                v2f ap;
                ap.x = pbuf[lg * 18 + kk * 4 + 2 * hi];
                ap.y = pbuf[lg * 18 + kk * 4 + 2 * hi + 1];
                v2f bv;
                bv.x = vl[(kk * 4 + 2 * hi) * DV + t * 16 + lg];
                bv.y = vl[(kk * 4 + 2 * hi + 1) * DV + t * 16 + lg];
                d = WMMA_F32X4(ap, bv, d);
            }
            acc[t] = d;
        }
        __syncthreads();
    }

    // normalize and store
    for (int t = 0; t < 4; ++t)
        for (int vg = 0; vg < 8; ++vg) {
            const int row = vg + 8 * hi;
            out[(size_t)(b * L + q0 + row) * DV + t * 16 + lg] = acc[t][vg] / l[vg];
        }
}

// ---------------------------------------------------------------------------
extern "C" void kernel_launch(void* const* d_in, const int* in_sizes, int n_in,
                              void* d_out, int out_size, void* d_ws, size_t ws_size,
                              hipStream_t stream) {
    using namespace cfg;
    const float* x    = (const float*)d_in[0];
    const float* Wq   = (const float*)d_in[1];
    const float* Wk   = (const float*)d_in[2];
    const float* Wv   = (const float*)d_in[3];
    const float* Wq_s = (const float*)d_in[4];
    const float* Wk_s = (const float*)d_in[5];
    const float* Wv_s = (const float*)d_in[6];
    const float* Wq_e = (const float*)d_in[7];
    const float* Wk_e = (const float*)d_in[8];
    const float* Wv_e = (const float*)d_in[9];
    const float* g_m  = (const float*)d_in[10];
    const float* b_m  = (const float*)d_in[11];
    const float* g_s  = (const float*)d_in[12];
    const float* b_s  = (const float*)d_in[13];
    const float* g_e  = (const float*)d_in[14];
    const float* b_e  = (const float*)d_in[15];
    const float* cope = (const float*)d_in[16];

    float* ws = (float*)d_ws;
    const size_t N = (size_t)B * L * DK;   // 589,824 floats per tensor
    float* qf = ws;
    float* kf = ws + N;
    float* vf = ws + 2 * N;

    ARC_ln_proj<<<(B * L) / 16, 128, 0, stream>>>(
        x, Wq, Wk, Wv, Wq_s, Wk_s, Wv_s, Wq_e, Wk_e, Wv_e,
        g_m, b_m, g_s, b_s, g_e, b_e, qf, kf, vf);

    ARC_cope_attn<<<B * NT, 32, 0, stream>>>(qf, kf, vf, cope, (float*)d_out);
}